// ARAPLoss_79912161509934
// MI455X (gfx1250) — compile-verified
//
#include <hip/hip_runtime.h>

#define TILE 256
#define NV_C 100000
#define B_C 8

// Truncate a generic (flat) pointer derived from a __shared__ array to its
// 32-bit LDS offset (flat LDS aperture keeps the LDS byte offset in addr[31:0]).
__device__ __forceinline__ unsigned lds_off_of(const void* p) {
    return (unsigned)(unsigned long long)p;
}

// CDNA5 async global->LDS copy, GVS mode: saddr(64b SGPR pair) + per-lane
// 32-bit byte offset. Tracked with ASYNCcnt.
__device__ __forceinline__ void async_b128(unsigned lds_addr, const int* sbase, unsigned voff) {
    asm volatile("global_load_async_to_lds_b128 %0, %1, %2"
                 :: "v"(lds_addr), "v"(voff), "s"(sbase)
                 : "memory");
}

extern "C" __global__ void arap_zero_kernel(float* __restrict__ out) {
    if (threadIdx.x < B_C) out[threadIdx.x] = 0.0f;
}

extern "C" __global__ void __launch_bounds__(TILE)
arap_partial_kernel(const float* __restrict__ dx, const float* __restrict__ x,
                    const int* __restrict__ esrc, const int* __restrict__ edst,
                    float* __restrict__ out, int E, int numTiles, float invE)
{
    __shared__ int   s_src[2][TILE];
    __shared__ int   s_dst[2][TILE];
    __shared__ float s_red[8][B_C];

    const int tid = threadIdx.x;

    float acc[B_C];
#pragma unroll
    for (int b = 0; b < B_C; ++b) acc[b] = 0.0f;

    const int step = gridDim.x;
    int buf = 0;

    // Stage one full tile of edge indices into LDS via the async unit.
    // Waves 0..3 (tid<128) each issue one b128 copy: 128 lanes * 16B = 2KB.
    // Returns whether the tile is full (uniform across the block).
    auto stage = [&](int t, int bsel) -> bool {
        bool full = ((t + 1) * TILE <= E);
        if (full && tid < 128) {
            const int* base = (tid < 64) ? esrc : edst;
            unsigned lane = (unsigned)(tid & 63);
            unsigned lds  = (tid < 64) ? lds_off_of(&s_src[bsel][0])
                                       : lds_off_of(&s_dst[bsel][0]);
            lds += lane * 16u;
            unsigned voff = ((unsigned)t * TILE + lane * 4u) * 4u; // byte offset
            async_b128(lds, base, voff);
        }
        return full;
    };

    bool cur_staged = false;
    if ((int)blockIdx.x < numTiles) cur_staged = stage((int)blockIdx.x, 0);

    for (int t = (int)blockIdx.x; t < numTiles; t += step) {
        const int tn = t + step;
        const bool next_staged = (tn < numTiles) ? stage(tn, buf ^ 1) : false;

        // Ensure the current tile's async copies have landed in LDS.
        // With the next tile already in flight we allow 1 outstanding.
        if (cur_staged) {
            if (next_staged) asm volatile("s_wait_asynccnt 0x1" ::: "memory");
            else             asm volatile("s_wait_asynccnt 0x0" ::: "memory");
        }
        __syncthreads();

        const int e = t * TILE + tid;
        int is = 0, id = 0;
        bool valid;
        if (cur_staged) {               // full tile: indices from LDS
            is = s_src[buf][tid];
            id = s_dst[buf][tid];
            valid = true;
        } else {                        // (at most one) partial tail tile
            valid = (e < E);
            if (valid) { is = esrc[e]; id = edst[e]; }
        }

        if (valid) {
            const unsigned os = 3u * (unsigned)is;
            const unsigned od = 3u * (unsigned)id;
#pragma unroll
            for (int b = 0; b < B_C; ++b) {
                const size_t ob = (size_t)b * (size_t)(NV_C * 3);
                const float* xs = x  + ob + os;   // 12B rows -> global_load_b96
                const float* xd = x  + ob + od;
                const float* ps = dx + ob + os;
                const float* pd = dx + ob + od;
                float ex0 = xd[0] - xs[0];
                float ex1 = xd[1] - xs[1];
                float ex2 = xd[2] - xs[2];
                float ed0 = pd[0] - ps[0];
                float ed1 = pd[1] - ps[1];
                float ed2 = pd[2] - ps[2];
                float dX = ex0 * ex0 + ex1 * ex1 + ex2 * ex2;
                float dD = ed0 * ed0 + ed1 * ed1 + ed2 * ed2;
                acc[b] += fabsf(dX - dD);
            }
        }

        __syncthreads();   // protect buf from being re-staged while still read
        buf ^= 1;
        cur_staged = next_staged;
    }

    // wave32 tree reduction, then cross-wave combine through LDS
    const int lane = tid & 31;
    const int wave = tid >> 5;
#pragma unroll
    for (int b = 0; b < B_C; ++b) {
        float v = acc[b];
#pragma unroll
        for (int off = 16; off > 0; off >>= 1) v += __shfl_down(v, off, 32);
        if (lane == 0) s_red[wave][b] = v;
    }
    __syncthreads();
    if (tid < B_C) {
        float s = 0.0f;
#pragma unroll
        for (int w = 0; w < 8; ++w) s += s_red[w][tid];
        unsafeAtomicAdd(&out[tid], s * invE);  // hardware global_atomic_add_f32
    }
}

extern "C" void kernel_launch(void* const* d_in, const int* in_sizes, int n_in,
                              void* d_out, int out_size, void* d_ws, size_t ws_size,
                              hipStream_t stream) {
    const float* dx  = (const float*)d_in[0];
    const float* x   = (const float*)d_in[1];
    const int* esrc  = (const int*)d_in[2];
    const int* edst  = (const int*)d_in[3];
    float* out = (float*)d_out;

    const int E = in_sizes[2];
    int numTiles = (E + TILE - 1) / TILE;
    if (numTiles < 1) numTiles = 1;
    int grid = numTiles < 2048 ? numTiles : 2048;
    const float invE = (E > 0) ? (1.0f / (float)E) : 0.0f;

    hipLaunchKernelGGL(arap_zero_kernel, dim3(1), dim3(32), 0, stream, out);
    hipLaunchKernelGGL(arap_partial_kernel, dim3(grid), dim3(TILE), 0, stream,
                       dx, x, esrc, edst, out, E, numTiles, invE);
}